// GaussianSlidingWindows_3667902071011
// MI455X (gfx1250) — compile-verified
//
#include <hip/hip_runtime.h>
#include <hip/hip_bf16.h>
#include <math.h>

// ---------------------------------------------------------------------------
// GMM attention (Graves) for MI455X / gfx1250, fp32 WMMA path + async LDS DMA.
// Sizes (fixed by reference): S=512, B=16, Din=512, Dfeat=512, K=10,
// T=NUM_CHAR=1024, Dctx=256.  Flattened row index rg = s*B + b in [0,8192).
// ---------------------------------------------------------------------------

typedef float v2f __attribute__((ext_vector_type(2)));
typedef float v8f __attribute__((ext_vector_type(8)));

#define SEQ      512
#define BATCH    16
#define ROWS     (SEQ * BATCH)      // 8192
#define DIN      512
#define DFEAT    512
#define NMIX     10
#define NOUT     30                 // 3*K
#define NCHAR    1024
#define DCTX     256
#define KC       64                 // K-chunk staged to LDS in ctx_kernel

static __device__ __forceinline__ v8f wmma_f32(v2f a, v2f b, v8f c) {
    return __builtin_amdgcn_wmma_f32_16x16x4_f32(
        /*neg_a=*/false, a, /*neg_b=*/false, b,
        /*c_mod=*/(short)0, c, /*reuse_a=*/false, /*reuse_b=*/false);
}

// ---------------------------------------------------------------------------
// Kernel A: fused 2-layer MLP + softplus + split/scale/clip.
// One block = one 16-row tile of the (8192 x 512) input; 4 waves.
// ---------------------------------------------------------------------------
__global__ __launch_bounds__(128) void mlp_kernel(
    const float* __restrict__ X,   // (8192, 512)
    const float* __restrict__ W1,  // (512, 512)
    const float* __restrict__ b1,  // (512)
    const float* __restrict__ W2,  // (512, 30)
    const float* __restrict__ b2,  // (30)
    float* __restrict__ out_vars,     // d_out vars section   (8192,10)
    float* __restrict__ out_weights,  // d_out weights section(8192,10)
    float* __restrict__ ws_delta,     // (8192,10)  delta_means/25
    float* __restrict__ ws_vars,      // (8192,10)  clipped vars
    float* __restrict__ ws_weights)   // (8192,10)
{
    __shared__ float sX[16 * DIN];   // 32 KB
    __shared__ float sH[16 * DFEAT]; // 32 KB

    const int m0   = blockIdx.x * 16;
    const int tid  = threadIdx.x;
    const int wave = tid >> 5;
    const int lane = tid & 31;
    const int mrow = lane & 15;          // M index inside tile
    const int koff = (lane >> 4) << 1;   // 0 or 2 : K-pair select (A/B frag layout)
    const int rsel = (lane >> 4) << 3;   // 0 or 8 : C/D row half

    // Stage X tile (16 x 512) in LDS, float4 vectorized.
    {
        const float4* Xv = reinterpret_cast<const float4*>(X + (size_t)m0 * DIN);
        float4* sXv = reinterpret_cast<float4*>(sX);
        #pragma unroll
        for (int i = tid; i < 16 * DIN / 4; i += 128) sXv[i] = Xv[i];
    }
    __syncthreads();

    // ---- GEMM1: h = leaky_relu(X @ W1 + b1); each wave owns 8 N-tiles ----
    for (int nt = wave; nt < DFEAT / 16; nt += 4) {
        const int ncol = nt * 16 + (lane & 15);
        v8f acc = {};
        #pragma unroll 4
        for (int k = 0; k < DIN; k += 4) {
            v2f a, b;
            a.x = sX[mrow * DIN + k + koff];
            a.y = sX[mrow * DIN + k + koff + 1];
            b.x = W1[(size_t)(k + koff    ) * DFEAT + ncol];
            b.y = W1[(size_t)(k + koff + 1) * DFEAT + ncol];
            acc = wmma_f32(a, b, acc);
        }
        const float bias = b1[ncol];
        #pragma unroll
        for (int r = 0; r < 8; ++r) {
            float v = acc[r] + bias;
            v = v > 0.0f ? v : 0.01f * v;           // leaky_relu(0.01)
            sH[(r + rsel) * DFEAT + ncol] = v;
        }
    }
    __syncthreads();

    // ---- GEMM2: outs = softplus(h @ W2 + b2); 30 cols -> waves 0,1 ----
    if (wave < 2) {
        const int ncol   = wave * 16 + (lane & 15);
        const bool valid = ncol < NOUT;
        v8f acc = {};
        #pragma unroll 4
        for (int k = 0; k < DFEAT; k += 4) {
            v2f a, b;
            a.x = sH[mrow * DFEAT + k + koff];
            a.y = sH[mrow * DFEAT + k + koff + 1];
            b.x = valid ? W2[(k + koff    ) * NOUT + ncol] : 0.0f;
            b.y = valid ? W2[(k + koff + 1) * NOUT + ncol] : 0.0f;
            acc = wmma_f32(a, b, acc);
        }
        if (valid) {
            const float bias = b2[ncol];
            #pragma unroll
            for (int r = 0; r < 8; ++r) {
                const int rg = m0 + r + rsel;       // flattened (s*B + b)
                float x  = acc[r] + bias;
                float sp = fmaxf(x, 0.0f) + log1pf(expf(-fabsf(x)));  // softplus
                if (ncol < NMIX) {
                    ws_delta[rg * NMIX + ncol] = sp * 0.04f;          // /25
                } else if (ncol < 2 * NMIX) {
                    const int kk = ncol - NMIX;
                    float v = fminf(fmaxf(sp, 0.01f), 100.0f);        // clip
                    ws_vars[rg * NMIX + kk]  = v;
                    out_vars[rg * NMIX + kk] = v;
                } else {
                    const int kk = ncol - 2 * NMIX;
                    ws_weights[rg * NMIX + kk]  = sp;
                    out_weights[rg * NMIX + kk] = sp;
                }
            }
        }
    }
}

// ---------------------------------------------------------------------------
// Kernel B: means = cumsum(delta, axis=S) + init_means. 160 series.
// ---------------------------------------------------------------------------
__global__ void cumsum_kernel(const float* __restrict__ ws_delta,
                              const float* __restrict__ init_means, // (16,10)
                              float* __restrict__ out_means,        // (8192,10)
                              float* __restrict__ ws_means)
{
    const int t = blockIdx.x * blockDim.x + threadIdx.x;
    if (t >= BATCH * NMIX) return;
    const int b = t / NMIX, k = t % NMIX;
    float acc = init_means[b * NMIX + k];
    for (int s = 0; s < SEQ; ++s) {
        const int idx = (s * BATCH + b) * NMIX + k;
        acc += ws_delta[idx];
        out_means[idx] = acc;
        ws_means[idx]  = acc;
    }
}

// ---------------------------------------------------------------------------
// Kernel C: attn[rg,t] = sum_k w_k * exp(-(mu_k - t)^2 * var_k).
// (exp(logsumexp(log w + d)) == the direct sum.)  One (s,b) row per block;
// each thread produces 4 consecutive t's -> one b128 store.
// ---------------------------------------------------------------------------
__global__ __launch_bounds__(256) void attn_kernel(
    const float* __restrict__ ws_means,
    const float* __restrict__ ws_vars,
    const float* __restrict__ ws_weights,
    float* __restrict__ out_attn)          // (8192, 1024)
{
    __shared__ float smu[NMIX], sva[NMIX], swt[NMIX];
    const int row = blockIdx.x;
    const int tid = threadIdx.x;
    if (tid < NMIX) {
        smu[tid] = ws_means[row * NMIX + tid];
        sva[tid] = ws_vars[row * NMIX + tid];
        swt[tid] = ws_weights[row * NMIX + tid];
    }
    __syncthreads();

    const float t0 = (float)(tid * 4);
    float a0 = 0.f, a1 = 0.f, a2 = 0.f, a3 = 0.f;
    #pragma unroll
    for (int k = 0; k < NMIX; ++k) {
        const float mu = smu[k], va = sva[k], w = swt[k];
        const float d0 = mu - t0;
        const float d1 = mu - (t0 + 1.0f);
        const float d2 = mu - (t0 + 2.0f);
        const float d3 = mu - (t0 + 3.0f);
        a0 += w * __expf(-d0 * d0 * va);
        a1 += w * __expf(-d1 * d1 * va);
        a2 += w * __expf(-d2 * d2 * va);
        a3 += w * __expf(-d3 * d3 * va);
    }
    float4 r = make_float4(a0, a1, a2, a3);
    reinterpret_cast<float4*>(out_attn + (size_t)row * NCHAR)[tid] = r;
}

// ---------------------------------------------------------------------------
// Kernel D: ctx_out[s,b,d] = sum_t attn[s,b,t] * ctx[t,b,d].
// Block (256 thr, 8 waves) owns one (b, n-tile, m-half); the 64x16 context
// chunk is DMA'd into LDS with global_load_async_to_lds_b128 (double
// buffered, overlapped via s_wait_asynccnt), each wave computes TWO 16x16
// output tiles per shared B fragment.
// ---------------------------------------------------------------------------
__global__ __launch_bounds__(256) void ctx_kernel(
    const float* __restrict__ AW,   // (8192, 1024), row = s*16 + b
    const float* __restrict__ CTX,  // (1024, 16, 256)
    float* __restrict__ out)        // (8192, 256)
{
    __shared__ float sB[2][KC * 16];           // 2 x 4 KB double buffer

    const int tid  = threadIdx.x;
    const int wave = tid >> 5;
    const int lane = tid & 31;
    const int gid  = blockIdx.x;               // 512 blocks: b(16) x nt(16) x half(2)
    const int b    = gid >> 5;
    const int nt   = (gid >> 1) & 15;
    const int half = gid & 1;
    const int mt0  = half * 16 + wave * 2;     // this wave's two m-tiles

    const int mrow   = lane & 15;
    const int koff   = (lane >> 4) << 1;       // 0/2: K-pair select
    const int rsel   = (lane >> 4) << 3;       // 0/8: C/D row half
    const int ncol16 = lane & 15;

    // Uniform global base of this block's B panel: CTX[t=0, b, nt*16].
    const float* bbase = CTX + (size_t)b * DCTX + nt * 16;
    // Per-thread staging slot: thread -> (t-row = tid>>2, d4 = tid&3) float4.
    const int trow = tid >> 2, d4 = tid & 3;
    // LDS byte address = low 32 bits of the flat address (ISA 10.2).
    const uint32_t lds0 = (uint32_t)(uintptr_t)(&sB[0][0]) + (uint32_t)tid * 16u;

    const float* arow0 = AW + ((size_t)((mt0 + 0) * 16 + mrow) * BATCH + b) * NCHAR;
    const float* arow1 = AW + ((size_t)((mt0 + 1) * 16 + mrow) * BATCH + b) * NCHAR;

    v8f acc0 = {}, acc1 = {};

    // Async-stage chunk kc (KC x 16 floats) into buffer kc&1.
    #define STAGE(kc)                                                           \
    {                                                                           \
        uint32_t voff = (uint32_t)(((kc) * KC + trow) * (BATCH * DCTX * 4))     \
                      + (uint32_t)(d4 * 16);                                    \
        uint32_t ldsa = lds0 + (uint32_t)(((kc) & 1) * (KC * 16 * 4));          \
        asm volatile("global_load_async_to_lds_b128 %0, %1, %2"                 \
                     :: "v"(ldsa), "v"(voff), "s"(bbase) : "memory");           \
    }

    STAGE(0);
    for (int kc = 0; kc < NCHAR / KC; ++kc) {
        if (kc + 1 < NCHAR / KC) {
            STAGE(kc + 1);                                   // prefetch next chunk
            asm volatile("s_wait_asynccnt 0x1" ::: "memory"); // chunk kc landed
        } else {
            asm volatile("s_wait_asynccnt 0x0" ::: "memory");
        }
        __syncthreads();                                     // visible to all waves

        const float* bb = &sB[kc & 1][0];
        const float* a0 = arow0 + kc * KC;
        const float* a1 = arow1 + kc * KC;
        #pragma unroll
        for (int k = 0; k < KC; k += 4) {
            v2f bf;
            bf.x = bb[(k + koff    ) * 16 + ncol16];
            bf.y = bb[(k + koff + 1) * 16 + ncol16];
            v2f af0 = *reinterpret_cast<const v2f*>(a0 + k + koff);
            v2f af1 = *reinterpret_cast<const v2f*>(a1 + k + koff);
            acc0 = wmma_f32(af0, bf, acc0);
            acc1 = wmma_f32(af1, bf, acc1);
        }
        __syncthreads();   // done reading this buffer before it is re-staged
    }
    #undef STAGE

    const int ncol = nt * 16 + ncol16;
    #pragma unroll
    for (int r = 0; r < 8; ++r) {
        out[((size_t)((mt0 + 0) * 16 + r + rsel) * BATCH + b) * DCTX + ncol] = acc0[r];
        out[((size_t)((mt0 + 1) * 16 + r + rsel) * BATCH + b) * DCTX + ncol] = acc1[r];
    }
}

// ---------------------------------------------------------------------------
extern "C" void kernel_launch(void* const* d_in, const int* in_sizes, int n_in,
                              void* d_out, int out_size, void* d_ws, size_t ws_size,
                              hipStream_t stream) {
    const float* X          = (const float*)d_in[0]; // (512,16,512)
    const float* CTX        = (const float*)d_in[1]; // (1024,16,256)
    const float* init_means = (const float*)d_in[2]; // (16,10)
    const float* W1         = (const float*)d_in[3];
    const float* b1         = (const float*)d_in[4];
    const float* W2         = (const float*)d_in[5];
    const float* b2         = (const float*)d_in[6];
    // d_in[7] = extra_chars == 0 (unused)

    float* out = (float*)d_out;
    float* out_ctx     = out;                                   // 512*16*256
    float* out_attn    = out + (size_t)ROWS * DCTX;             // 512*16*1024
    float* out_means   = out_attn + (size_t)ROWS * NCHAR;       // 512*16*10
    float* out_vars    = out_means + (size_t)ROWS * NMIX;
    float* out_weights = out_vars  + (size_t)ROWS * NMIX;

    float* ws         = (float*)d_ws;
    float* ws_delta   = ws;                       // 81920 floats each
    float* ws_vars    = ws + 1 * ROWS * NMIX;
    float* ws_weights = ws + 2 * ROWS * NMIX;
    float* ws_means   = ws + 3 * ROWS * NMIX;     // total 1.31 MB

    mlp_kernel<<<ROWS / 16, 128, 0, stream>>>(X, W1, b1, W2, b2,
                                              out_vars, out_weights,
                                              ws_delta, ws_vars, ws_weights);
    cumsum_kernel<<<1, BATCH * NMIX, 0, stream>>>(ws_delta, init_means,
                                                  out_means, ws_means);
    attn_kernel<<<ROWS, 256, 0, stream>>>(ws_means, ws_vars, ws_weights, out_attn);
    ctx_kernel<<<16 * 16 * 2, 256, 0, stream>>>(out_attn, CTX, out_ctx);
}